// DifferentiableNeuralDict_16527034155821
// MI455X (gfx1250) — compile-verified
//
#include <hip/hip_runtime.h>

typedef float v2f __attribute__((ext_vector_type(2)));
typedef float v8f __attribute__((ext_vector_type(8)));

#define DIMS   64
#define NNEI   50
#define BATCH  4096
#define DELTA  1e-3f

// One wave32 per batch row. 4 tiles of 16 neighbors, chunk-outer loop.
// Squared distances computed as diagonal of diff*diff^T via
// V_WMMA_F32_16X16X4_F32 (A and B share the same register pair -> A*A^T).
__global__ __launch_bounds__(256) void dnd_wmma_kernel(
    const float*     __restrict__ key,           // [B, 64]
    const long long* __restrict__ indices,       // [B, 50] int64
    const float*     __restrict__ keys_table,    // [C, 64]
    const float*     __restrict__ values_table,  // [C]
    float*           __restrict__ out)           // [B]
{
    const int lane = threadIdx.x & 31;
    const int wave = threadIdx.x >> 5;
    const int row  = blockIdx.x * 8 + wave;

    const int hi = lane >> 4;     // 0: lanes 0-15, 1: lanes 16-31
    const int m  = lane & 15;     // neighbor slot within a 16-tile

    // Query preload: lane j holds dims (2j, 2j+1). Fully coalesced 256B.
    v2f qreg = *(const v2f*)(key + (size_t)row * DIMS + 2 * lane);

    const long long* idxrow = indices + (size_t)row * NNEI;

    // All 4 tiles' indices up front; lanes l and l+16 share a neighbor row.
    long long idx4[4];
#pragma unroll
    for (int t = 0; t < 4; ++t) {
        int n = t * 16 + m;
        idx4[t] = (n < NNEI) ? idxrow[n] : 0;   // idx 0 is safe padding
    }

    // Row base pointers: lane l<16 reads dims (k0,k0+1), l>=16 (k0+2,k0+3).
    const float* rowp[4];
#pragma unroll
    for (int t = 0; t < 4; ++t)
        rowp[t] = keys_table + (size_t)idx4[t] * DIMS + 2 * hi;

    // Kick off line fetches (256B row = 2 lines; lane halves cover both).
#pragma unroll
    for (int t = 0; t < 4; ++t)
        __builtin_prefetch(keys_table + (size_t)idx4[t] * DIMS + hi * 32, 0, 1);

    // 4 independent accumulators -> no WMMA RAW chain inside a chunk.
    v8f acc[4];
#pragma unroll
    for (int t = 0; t < 4; ++t) acc[t] = (v8f){};

#pragma unroll
    for (int k0 = 0; k0 < DIMS; k0 += 4) {
        // Query dims for this lane's (k0,hi) live in lane k0/2 + hi.
        // Fetched ONCE per chunk, shared by all 4 tiles.
        int   srcLane = (k0 >> 1) + hi;
        float qx = __shfl(qreg.x, srcLane, 32);
        float qy = __shfl(qreg.y, srcLane, 32);
#pragma unroll
        for (int t = 0; t < 4; ++t) {
            v2f cv = *(const v2f*)(rowp[t] + k0);   // 8B-aligned gather
            v2f a;
            a.x = cv.x - qx;
            a.y = cv.y - qy;
            // D_t += A * A^T : diagonal accumulates squared distances
            acc[t] = __builtin_amdgcn_wmma_f32_16x16x4_f32(
                false, a, false, a, (short)0, acc[t], false, false);
        }
    }

    float wsum = 0.0f, wvsum = 0.0f;

#pragma unroll
    for (int t = 0; t < 4; ++t) {
        // Diagonal: D[r][r] at (VGPR r, lane r); D[8+r][8+r] at (VGPR r,
        // lane 24+r). Lane indices are unroll-constants -> v_readlane.
        float diag = 0.0f;
#pragma unroll
        for (int r = 0; r < 8; ++r) {
            float dlo = __int_as_float(
                __builtin_amdgcn_readlane(__float_as_int(acc[t][r]), r));
            float dhi = __int_as_float(
                __builtin_amdgcn_readlane(__float_as_int(acc[t][r]), 24 + r));
            diag = (lane == r)     ? dlo : diag;
            diag = (lane == 8 + r) ? dhi : diag;
        }

        int   n = t * 16 + m;
        float v = values_table[idx4[t]];
        bool  active = (hi == 0) && (n < NNEI);   // lanes 0-15, valid neighbors
        float w = active ? (1.0f / (diag + DELTA)) : 0.0f;
        wsum  += w;
        wvsum += w * v;
    }

    // Wave reduction (nonzero contributions only in lanes 0-15).
#pragma unroll
    for (int off = 16; off >= 1; off >>= 1) {
        wsum  += __shfl_xor(wsum,  off, 32);
        wvsum += __shfl_xor(wvsum, off, 32);
    }
    if (lane == 0) out[row] = wvsum / wsum;
}

extern "C" void kernel_launch(void* const* d_in, const int* in_sizes, int n_in,
                              void* d_out, int out_size, void* d_ws, size_t ws_size,
                              hipStream_t stream) {
    const float*     key          = (const float*)d_in[0];
    const long long* indices      = (const long long*)d_in[1];
    const float*     keys_table   = (const float*)d_in[2];
    const float*     values_table = (const float*)d_in[3];
    float*           out          = (float*)d_out;

    // 4096 rows, 8 waves (rows) per 256-thread block -> 512 blocks.
    dnd_wmma_kernel<<<BATCH / 8, 256, 0, stream>>>(
        key, indices, keys_table, values_table, out);
}